// GCNClassifier_79998060855858
// MI455X (gfx1250) — compile-verified
//
#include <hip/hip_runtime.h>
#include <hip/hip_bf16.h>

typedef __attribute__((ext_vector_type(2))) float v2f;
typedef __attribute__((ext_vector_type(8))) float v8f;

#define EPSV 1e-5f

// ---------------------------------------------------------------------------
// GEMM: H[N,128] = X[N,128] @ W[128,128] using V_WMMA_F32_16X16X4_F32.
// One wave computes one 16x16 output tile; block = 8 waves = 8 N-tiles (full
// 128-wide output row of tiles); grid.x = M-tiles.
// A layout (ISA 7.12.2, 32-bit A 16x4): lanes0-15 M=0..15, VGPR0={K0|K2},
// VGPR1={K1|K3} (halves of the wave hold K+0/1 vs K+2/3).
// C/D layout: VGPR v: lanes0-15 -> (M=v, N=lane), lanes16-31 -> (M=v+8).
// ---------------------------------------------------------------------------
__global__ __launch_bounds__(256) void gcn_gemm_wmma(const float* __restrict__ X,
                                                     const float* __restrict__ W,
                                                     float* __restrict__ H,
                                                     int nrows) {
  const int mtile = blockIdx.x;
  const int wave  = threadIdx.x >> 5;     // 0..7 -> N tile
  const int lane  = threadIdx.x & 31;
  const int half  = lane >> 4;            // 0: K+0/1, 1: K+2/3
  const int l     = lane & 15;
  const int m0 = mtile * 16;
  const int n0 = wave * 16;

  int row = m0 + l;
  if (row >= nrows) row = nrows - 1;      // keep EXEC all-ones for WMMA

  v8f c = {};
  const float* xr = X + (size_t)row * 128;
  for (int k0 = 0; k0 < 128; k0 += 4) {
    const int ka = k0 + 2 * half;
    v2f a, b;
    a.x = xr[ka];
    a.y = xr[ka + 1];
    b.x = W[ka * 128 + n0 + l];
    b.y = W[(ka + 1) * 128 + n0 + l];
    c = __builtin_amdgcn_wmma_f32_16x16x4_f32(false, a, false, b,
                                              (short)0, c, false, false);
  }
#pragma unroll
  for (int v = 0; v < 8; ++v) {
    const int r = m0 + v + half * 8;
    if (r < nrows) H[(size_t)r * 128 + n0 + l] = c[v];
  }
}

// ---------------------------------------------------------------------------
// Degree accumulation over edge destinations (self-loop folded in later).
// ---------------------------------------------------------------------------
__global__ __launch_bounds__(256) void deg_accum(const int* __restrict__ dst,
                                                 int E, float* __restrict__ deg) {
  int e = blockIdx.x * 256 + threadIdx.x;
  if (e < E) atomicAdd(&deg[dst[e]], 1.0f);
}

__global__ __launch_bounds__(256) void dinv_finalize(float* __restrict__ deg, int n) {
  int i = blockIdx.x * 256 + threadIdx.x;
  if (i < n) deg[i] = rsqrtf(deg[i] + 1.0f);   // +1: self-loop
}

// ---------------------------------------------------------------------------
// Edge aggregation: wave per edge. 512B coalesced row gather (L2 resident)
// + 128 contiguous fire-and-forget f32 atomics per edge.
// ---------------------------------------------------------------------------
__global__ __launch_bounds__(256) void agg_edges(const float* __restrict__ H,
                                                 const int* __restrict__ src,
                                                 const int* __restrict__ dst,
                                                 const float* __restrict__ dinv,
                                                 float* __restrict__ out, int E) {
  const int tid  = blockIdx.x * 256 + threadIdx.x;
  const int e    = tid >> 5;
  const int lane = tid & 31;
  if (e >= E) return;
  const int s = src[e];
  const int d = dst[e];
  const float w = dinv[s] * dinv[d];
  const float4 v = *(const float4*)(H + (size_t)s * 128 + lane * 4);
  float* o = out + (size_t)d * 128 + lane * 4;
  atomicAdd(o + 0, v.x * w);
  atomicAdd(o + 1, v.y * w);
  atomicAdd(o + 2, v.z * w);
  atomicAdd(o + 3, v.w * w);
}

// self-loop contribution + bias (in place on aggregation buffer)
__global__ __launch_bounds__(256) void finalize_self_bias(const float* __restrict__ H,
                                                          const float* __restrict__ dinv,
                                                          const float* __restrict__ bias,
                                                          float* __restrict__ out,
                                                          int total) {
  int idx = blockIdx.x * 256 + threadIdx.x;
  if (idx >= total) return;
  const int i = idx >> 7;
  const int c = idx & 127;
  const float di = dinv[i];
  out[idx] += H[idx] * di * di + bias[c];
}

// ---------------------------------------------------------------------------
// Column mean/var (population) over nodes: one block per column.
// ---------------------------------------------------------------------------
__global__ __launch_bounds__(256) void col_stats(const float* __restrict__ X, int n,
                                                 float* __restrict__ mean,
                                                 float* __restrict__ var) {
  __shared__ float ssum[256];
  __shared__ float ssq[256];
  const int c = blockIdx.x;
  const int t = threadIdx.x;
  float s = 0.f, q = 0.f;
  for (int r = t; r < n; r += 256) {
    const float v = X[(size_t)r * 128 + c];
    s += v; q += v * v;
  }
  ssum[t] = s; ssq[t] = q;
  __syncthreads();
  for (int off = 128; off > 0; off >>= 1) {
    if (t < off) { ssum[t] += ssum[t + off]; ssq[t] += ssq[t + off]; }
    __syncthreads();
  }
  if (t == 0) {
    const float mu = ssum[0] / (float)n;
    mean[c] = mu;
    var[c]  = ssq[0] / (float)n - mu * mu;
  }
}

__global__ __launch_bounds__(256) void bn_apply(float* __restrict__ X,
                                                const float* __restrict__ mean,
                                                const float* __restrict__ var,
                                                const float* __restrict__ g,
                                                const float* __restrict__ be,
                                                int total, int relu) {
  int idx = blockIdx.x * 256 + threadIdx.x;
  if (idx >= total) return;
  const int c = idx & 127;
  float y = (X[idx] - mean[c]) * rsqrtf(var[c] + EPSV) * g[c] + be[c];
  if (relu) y = fmaxf(y, 0.f);
  X[idx] = y;
}

// ---------------------------------------------------------------------------
// Global mean pool: wave per node, atomic accumulate into [G,128] sums.
// ---------------------------------------------------------------------------
__global__ __launch_bounds__(256) void pool_accum(const float* __restrict__ X,
                                                  const int* __restrict__ batch,
                                                  float* __restrict__ gsum,
                                                  float* __restrict__ gcnt, int n) {
  const int tid  = blockIdx.x * 256 + threadIdx.x;
  const int i    = tid >> 5;
  const int lane = tid & 31;
  if (i >= n) return;
  const int g = batch[i];
  const float4 v = *(const float4*)(X + (size_t)i * 128 + lane * 4);
  float* o = gsum + (size_t)g * 128 + lane * 4;
  atomicAdd(o + 0, v.x);
  atomicAdd(o + 1, v.y);
  atomicAdd(o + 2, v.z);
  atomicAdd(o + 3, v.w);
  if (lane == 0) atomicAdd(&gcnt[g], 1.0f);
}

// ---------------------------------------------------------------------------
// MLP head: block per graph. z = relu(gemb @ Wc1 + bc1); logits = z @ Wc2 + bc2
// ---------------------------------------------------------------------------
__global__ __launch_bounds__(64) void mlp_head(const float* __restrict__ gsum,
                                               const float* __restrict__ gcnt,
                                               const float* __restrict__ Wc1,
                                               const float* __restrict__ bc1,
                                               const float* __restrict__ Wc2,
                                               const float* __restrict__ bc2,
                                               float* __restrict__ out) {
  __shared__ float emb[128];
  __shared__ float z[64];
  const int g = blockIdx.x;
  const int j = threadIdx.x;
  const float inv = 1.0f / fmaxf(gcnt[g], 1.0f);
  emb[j]      = gsum[(size_t)g * 128 + j] * inv;
  emb[j + 64] = gsum[(size_t)g * 128 + 64 + j] * inv;
  __syncthreads();
  float acc = bc1[j];
  for (int k = 0; k < 128; ++k) acc += emb[k] * Wc1[k * 64 + j];
  z[j] = fmaxf(acc, 0.f);
  __syncthreads();
  if (j < 2) {
    float o = bc2[j];
    for (int k = 0; k < 64; ++k) o += z[k] * Wc2[k * 2 + j];
    out[g * 2 + j] = o;
  }
}

// ---------------------------------------------------------------------------
extern "C" void kernel_launch(void* const* d_in, const int* in_sizes, int n_in,
                              void* d_out, int out_size, void* d_ws, size_t ws_size,
                              hipStream_t stream) {
  const float* x    = (const float*)d_in[0];
  const int*   eidx = (const int*)d_in[1];
  const int*   batch= (const int*)d_in[2];
  const float* W1   = (const float*)d_in[3];
  const float* b1   = (const float*)d_in[4];
  const float* g1   = (const float*)d_in[5];
  const float* be1  = (const float*)d_in[6];
  const float* W2   = (const float*)d_in[7];
  const float* b2   = (const float*)d_in[8];
  const float* g2   = (const float*)d_in[9];
  const float* be2  = (const float*)d_in[10];
  const float* W3   = (const float*)d_in[11];
  const float* b3   = (const float*)d_in[12];
  const float* g3   = (const float*)d_in[13];
  const float* be3  = (const float*)d_in[14];
  const float* Wc1  = (const float*)d_in[15];
  const float* bc1  = (const float*)d_in[16];
  const float* Wc2  = (const float*)d_in[17];
  const float* bc2  = (const float*)d_in[18];

  const int n = in_sizes[0] / 128;        // 50000 nodes
  const int E = in_sizes[1] / 2;          // 1.6M edges
  const int* src = eidx;
  const int* dst = eidx + E;

  float* ws = (float*)d_ws;
  const size_t nf = (size_t)n * 128;
  float* H    = ws;                        // GEMM output
  float* P    = ws + nf;                   // activation ping
  float* Q    = ws + 2 * nf;               // activation pong
  float* dinv = ws + 3 * nf;               // [n]
  float* mean = dinv + n;                  // [128]
  float* var  = mean + 128;                // [128]
  float* gsum = var + 128;                 // [64*128]
  float* gcnt = gsum + 64 * 128;           // [64]

  const int mtiles    = (n + 15) / 16;
  const int aggBlocks = (int)(((long long)E * 32 + 255) / 256);
  const int ewBlocks  = (int)((nf + 255) / 256);
  const int poolBlks  = (int)(((long long)n * 32 + 255) / 256);
  const int total     = (int)nf;

  // symmetric-normalization degrees (once; self-loop folded into +1)
  hipMemsetAsync(dinv, 0, (size_t)n * sizeof(float), stream);
  deg_accum<<<(E + 255) / 256, 256, 0, stream>>>(dst, E, dinv);
  dinv_finalize<<<(n + 255) / 256, 256, 0, stream>>>(dinv, n);

  auto layer = [&](const float* in, const float* W, const float* b,
                   const float* g, const float* be, float* out, int relu) {
    gcn_gemm_wmma<<<mtiles, 256, 0, stream>>>(in, W, H, n);
    hipMemsetAsync(out, 0, nf * sizeof(float), stream);
    agg_edges<<<aggBlocks, 256, 0, stream>>>(H, src, dst, dinv, out, E);
    finalize_self_bias<<<ewBlocks, 256, 0, stream>>>(H, dinv, b, out, total);
    col_stats<<<128, 256, 0, stream>>>(out, n, mean, var);
    bn_apply<<<ewBlocks, 256, 0, stream>>>(out, mean, var, g, be, total, relu);
  };

  layer(x, W1, b1, g1, be1, P, 1);
  layer(P, W2, b2, g2, be2, Q, 1);
  layer(Q, W3, b3, g3, be3, P, 0);

  hipMemsetAsync(gsum, 0, (64 * 128 + 64) * sizeof(float), stream);
  pool_accum<<<poolBlks, 256, 0, stream>>>(P, batch, gsum, gcnt, n);
  mlp_head<<<64, 64, 0, stream>>>(gsum, gcnt, Wc1, bc1, Wc2, bc2, (float*)d_out);
}